// GraphEmbeddingWithSoftPrompt_71854802862715
// MI455X (gfx1250) — compile-verified
//
#include <hip/hip_runtime.h>
#include <hip/hip_bf16.h>

// ---- Problem constants (from reference) ----
#define V_SIZE    30522
#define H_DIM     768
#define B_SIZE    64
#define S_LEN     512
#define NEW_ROWS  10004
#define N_NODES   10002      // NEW_ROWS - 2
#define N_SOFT    8
#define N_EDGES   50000
#define LEAKY     0.2f

typedef __attribute__((ext_vector_type(2))) float v2f;
typedef __attribute__((ext_vector_type(8))) float v8f;

// ---------------------------------------------------------------------------
// 0) init: ef[0..N_NODES-1] = new_weight[1+i] (residual), ef[N_NODES] = new_weight[-1]
//    mbits = 0 (== -inf in ordered-uint encoding), denom = 0
// ---------------------------------------------------------------------------
__global__ __launch_bounds__(256) void init_ws_kernel(
    const float* __restrict__ new_weight, float* __restrict__ ef,
    unsigned* __restrict__ mbits, float* __restrict__ denom)
{
    size_t i = (size_t)blockIdx.x * blockDim.x + threadIdx.x;
    const size_t total = (size_t)(N_NODES + 1) * H_DIM;
    if (i < total) {
        size_t r = i / H_DIM, c = i - r * H_DIM;
        size_t srow = (r < N_NODES) ? (r + 1) : (size_t)(NEW_ROWS - 1);
        ef[i] = new_weight[srow * H_DIM + c];
    }
    if (i < N_NODES) { mbits[i] = 0u; denom[i] = 0.f; }
}

// ---------------------------------------------------------------------------
// 0b) Wt[n][k] = W[k][n]  (LDS-tiled transpose of the 768x768 GAT weight so
//     B fragments become contiguous b64 loads in the WMMA loop)
// ---------------------------------------------------------------------------
__global__ __launch_bounds__(256) void transpose_w_kernel(
    const float* __restrict__ W, float* __restrict__ Wt)
{
    __shared__ float tile[32][33];
    const int TB = H_DIM / 32;                  // 24 tiles per side
    int bx = blockIdx.x % TB, by = blockIdx.x / TB;
    int tx = threadIdx.x & 31, ty = threadIdx.x >> 5;   // 32x8 threads
    #pragma unroll
    for (int i = 0; i < 32; i += 8)
        tile[ty + i][tx] = W[(size_t)(by * 32 + ty + i) * H_DIM + bx * 32 + tx];
    __syncthreads();
    #pragma unroll
    for (int i = 0; i < 32; i += 8)
        Wt[(size_t)(bx * 32 + ty + i) * H_DIM + by * 32 + tx] = tile[tx][ty + i];
}

// ---------------------------------------------------------------------------
// 1) GEMM: h[10002,768] = new_weight[1:10003] @ W_gat, f32 WMMA 16x16x4.
//    One wave owns a 16x64 output strip (4 N-tiles share one A fragment).
//    A 16x4 layout (ISA 7.12.2): lanes 0-15 hold K={0,1}, lanes 16-31 K={2,3}.
//    C/D layout: VGPR i -> M=i (lanes 0-15) / M=8+i (lanes 16-31), N=lane&15.
//    Boundary M-tile handled by clamped-row load * 0/1 mask (no exec branches).
// ---------------------------------------------------------------------------
__global__ __launch_bounds__(128) void gat_gemm_wmma_kernel(
    const float* __restrict__ A,    // new_weight + H_DIM  (rows 1..10002)
    const float* __restrict__ Wt,   // W_gat transposed [768,768], Wt[n][k]
    float* __restrict__ Hout)       // [10002,768]
{
    const int lane = threadIdx.x & 31;
    const int wid  = blockIdx.x * 4 + (threadIdx.x >> 5);
    const int NG   = H_DIM / 64;                 // 12 strips of 64 cols
    const int mt   = wid / NG;
    const int ng   = wid % NG;
    if (mt >= (N_NODES + 15) / 16) return;       // 626 M-tiles
    const int m0 = mt * 16;
    const int n0 = ng * 64;

    const int   lhalf = lane & 15;
    const int   khi   = (lane >> 4) << 1;        // 0 (lanes 0-15) or 2 (16-31)
    const int   arow  = m0 + lhalf;
    const float amask = (arow < N_NODES) ? 1.f : 0.f;
    const int   crow  = (arow < N_NODES) ? arow : (N_NODES - 1);  // clamped
    const float* arp  = A  + (size_t)crow * H_DIM + khi;
    const float* wtp  = Wt + (size_t)(n0 + lhalf) * H_DIM + khi;

    v8f c0 = {}, c1 = {}, c2 = {}, c3 = {};
    for (int k0 = 0; k0 < H_DIM; k0 += 4) {
        v2f a = *(const v2f*)(arp + k0);
        a.x *= amask;
        a.y *= amask;
        v2f b0 = *(const v2f*)(wtp + k0);
        v2f b1 = *(const v2f*)(wtp + 16 * H_DIM + k0);
        v2f b2 = *(const v2f*)(wtp + 32 * H_DIM + k0);
        v2f b3 = *(const v2f*)(wtp + 48 * H_DIM + k0);
        c0 = __builtin_amdgcn_wmma_f32_16x16x4_f32(false, a, false, b0, (short)0, c0, false, false);
        c1 = __builtin_amdgcn_wmma_f32_16x16x4_f32(false, a, false, b1, (short)0, c1, false, false);
        c2 = __builtin_amdgcn_wmma_f32_16x16x4_f32(false, a, false, b2, (short)0, c2, false, false);
        c3 = __builtin_amdgcn_wmma_f32_16x16x4_f32(false, a, false, b3, (short)0, c3, false, false);
    }

    const int rbase = m0 + ((lane >> 4) << 3);   // +8 for lanes 16-31
    #pragma unroll
    for (int i = 0; i < 8; ++i) {
        int r = rbase + i;
        if (r < N_NODES) {
            float* orow = Hout + (size_t)r * H_DIM + n0 + lhalf;
            orow[0]  = c0[i];
            orow[16] = c1[i];
            orow[32] = c2[i];
            orow[48] = c3[i];
        }
    }
}

// ---------------------------------------------------------------------------
// 2) s_src[r] = h[r]·a_src ; s_dst[r] = h[r]·a_dst  (one wave32 per row)
// ---------------------------------------------------------------------------
__global__ __launch_bounds__(128) void row_dots_kernel(
    const float* __restrict__ Hm, const float* __restrict__ a_src,
    const float* __restrict__ a_dst, float* __restrict__ s_src,
    float* __restrict__ s_dst)
{
    int wave = (blockIdx.x * blockDim.x + threadIdx.x) >> 5;
    int lane = threadIdx.x & 31;
    if (wave >= N_NODES) return;
    const float* row = Hm + (size_t)wave * H_DIM;
    float ss = 0.f, sd = 0.f;
    for (int c = lane; c < H_DIM; c += 32) {
        float v = row[c];
        ss += v * a_src[c];
        sd += v * a_dst[c];
    }
    #pragma unroll
    for (int off = 16; off > 0; off >>= 1) {
        ss += __shfl_down(ss, off, 32);
        sd += __shfl_down(sd, off, 32);
    }
    if (lane == 0) { s_src[wave] = ss; s_dst[wave] = sd; }
}

// ordered-uint encoding of f32 so unsigned atomicMax == float max
__device__ __forceinline__ unsigned fenc(float f) {
    unsigned u = __float_as_uint(f);
    return (u & 0x80000000u) ? ~u : (u | 0x80000000u);
}
__device__ __forceinline__ float fdec(unsigned u) {
    unsigned b = (u & 0x80000000u) ? (u & 0x7FFFFFFFu) : ~u;
    return __uint_as_float(b);
}

// ---------------------------------------------------------------------------
// 3) e = leaky_relu(s_src[src] + s_dst[dst]); segment max over dst
// ---------------------------------------------------------------------------
__global__ __launch_bounds__(256) void edge_e_max_kernel(
    const int* __restrict__ ei, const float* __restrict__ s_src,
    const float* __restrict__ s_dst, float* __restrict__ e,
    unsigned* __restrict__ mbits)
{
    int t = blockIdx.x * blockDim.x + threadIdx.x;
    if (t >= N_EDGES) return;
    int s = ei[t], d = ei[N_EDGES + t];
    float v = s_src[s] + s_dst[d];
    v = (v >= 0.f) ? v : LEAKY * v;
    e[t] = v;
    atomicMax(&mbits[d], fenc(v));
}

// ---------------------------------------------------------------------------
// 4) ex = exp(e - m[dst]); denom[dst] += ex
// ---------------------------------------------------------------------------
__global__ __launch_bounds__(256) void edge_exp_sum_kernel(
    const int* __restrict__ ei, const float* __restrict__ e,
    const unsigned* __restrict__ mbits, float* __restrict__ ex,
    float* __restrict__ denom)
{
    int t = blockIdx.x * blockDim.x + threadIdx.x;
    if (t >= N_EDGES) return;
    int d = ei[N_EDGES + t];
    float v = __expf(e[t] - fdec(mbits[d]));
    ex[t] = v;
    atomicAdd(&denom[d], v);
}

// ---------------------------------------------------------------------------
// 5) ef[dst] += alpha * h[src]   (one block per edge; ef pre-loaded with feat)
// ---------------------------------------------------------------------------
__global__ __launch_bounds__(256) void edge_scatter_kernel(
    const int* __restrict__ ei, const float* __restrict__ ex,
    const float* __restrict__ denom, const float* __restrict__ Hm,
    float* __restrict__ ef)
{
    int eix = blockIdx.x;
    int s = ei[eix], d = ei[N_EDGES + eix];
    float alpha = ex[eix] / (denom[d] + 1e-16f);
    const float* hrow = Hm + (size_t)s * H_DIM;
    float*       erow = ef + (size_t)d * H_DIM;
    for (int c = threadIdx.x; c < H_DIM; c += 256)
        atomicAdd(&erow[c], alpha * hrow[c]);
}

// ---------------------------------------------------------------------------
// 6) Output gather: one 192-thread block per (b,s) row, float4 per thread.
// ---------------------------------------------------------------------------
__global__ __launch_bounds__(192) void gather_out_kernel(
    const int* __restrict__ x, const float* __restrict__ orig,
    const float* __restrict__ ef, const float* __restrict__ soft,
    float* __restrict__ out)
{
    int row = blockIdx.x;           // 0 .. B*S-1
    int b = row / S_LEN, s = row - b * S_LEN;
    const float4* src;
    if (s < N_SOFT) {
        src = (const float4*)(soft + (size_t)s * H_DIM);
    } else {
        int idx = x[b * S_LEN + s];
        src = (idx < V_SIZE) ? (const float4*)(orig + (size_t)idx * H_DIM)
                             : (const float4*)(ef + (size_t)(idx - V_SIZE) * H_DIM);
    }
    float4* dst = (float4*)(out + (size_t)row * H_DIM);
    dst[threadIdx.x] = src[threadIdx.x];
}

// ---------------------------------------------------------------------------
extern "C" void kernel_launch(void* const* d_in, const int* in_sizes, int n_in,
                              void* d_out, int out_size, void* d_ws, size_t ws_size,
                              hipStream_t stream) {
    const int*   x           = (const int*)d_in[0];
    const int*   edge_index  = (const int*)d_in[1];
    const float* orig_w      = (const float*)d_in[2];
    const float* new_w       = (const float*)d_in[3];
    const float* soft_w      = (const float*)d_in[4];
    const float* W_gat       = (const float*)d_in[5];
    const float* a_src       = (const float*)d_in[6];
    const float* a_dst       = (const float*)d_in[7];
    float* out = (float*)d_out;

    // workspace bump allocator (256B aligned)
    char* ws = (char*)d_ws;
    auto take = [&](size_t bytes) -> char* {
        char* p = ws;
        ws += (bytes + 255) & ~(size_t)255;
        return p;
    };
    float*    h     = (float*)   take((size_t)N_NODES * H_DIM * 4);       // ~30.7 MB
    float*    ef    = (float*)   take((size_t)(N_NODES + 1) * H_DIM * 4); // ~30.7 MB
    float*    Wt    = (float*)   take((size_t)H_DIM * H_DIM * 4);         // ~2.4 MB
    float*    ssrc  = (float*)   take((size_t)N_NODES * 4);
    float*    sdst  = (float*)   take((size_t)N_NODES * 4);
    unsigned* mbits = (unsigned*)take((size_t)N_NODES * 4);
    float*    denom = (float*)   take((size_t)N_NODES * 4);
    float*    e     = (float*)   take((size_t)N_EDGES * 4);
    float*    ex    = (float*)   take((size_t)N_EDGES * 4);
    (void)ws_size; (void)in_sizes; (void)n_in; (void)out_size;

    // 0) init residual table + segment state (re-done every call: determinism)
    {
        size_t total = (size_t)(N_NODES + 1) * H_DIM;
        int blocks = (int)((total + 255) / 256);
        init_ws_kernel<<<blocks, 256, 0, stream>>>(new_w, ef, mbits, denom);
    }
    // 0b) transpose W_gat for contiguous B-fragment loads
    {
        int TB = H_DIM / 32;
        transpose_w_kernel<<<TB * TB, 256, 0, stream>>>(W_gat, Wt);
    }
    // 1) WMMA GEMM  h = feat @ W_gat
    {
        int mtiles = (N_NODES + 15) / 16;        // 626
        int waves  = mtiles * (H_DIM / 64);      // 626*12 = 7512
        gat_gemm_wmma_kernel<<<waves / 4, 128, 0, stream>>>(new_w + H_DIM, Wt, h);
    }
    // 2) attention scalar dots
    row_dots_kernel<<<(N_NODES + 3) / 4, 128, 0, stream>>>(h, a_src, a_dst, ssrc, sdst);
    // 3) edge scores + segment max
    edge_e_max_kernel<<<(N_EDGES + 255) / 256, 256, 0, stream>>>(edge_index, ssrc, sdst, e, mbits);
    // 4) exp + segment sum
    edge_exp_sum_kernel<<<(N_EDGES + 255) / 256, 256, 0, stream>>>(edge_index, e, mbits, ex, denom);
    // 5) weighted scatter-aggregate into residual table
    edge_scatter_kernel<<<N_EDGES, 256, 0, stream>>>(edge_index, ex, denom, h, ef);
    // 6) final embedding gather (+ soft prompt rows)
    gather_out_kernel<<<B_SIZE * S_LEN, 192, 0, stream>>>(x, orig_w, ef, soft_w, out);
}